// VectorQuantizer_42494406427019
// MI455X (gfx1250) — compile-verified
//
#include <hip/hip_runtime.h>

// VQ quantizer for MI455X (gfx1250, wave32).
// z: [16, 256, 32, 32] f32, emb: [1024, 256] f32.
// out: z_q [16,256,32,32] (= emb[argmin]) followed by scalar loss.

#define D_DIM   256
#define K_CODES 1024
#define N_VECS  16384          // 16 * 32 * 32
#define BATCH_STRIDE 262144    // 256 * 1024 floats per batch image

typedef float v2f __attribute__((ext_vector_type(2)));
typedef float v8f __attribute__((ext_vector_type(8)));

// ---------------------------------------------------------------------------
// Kernel 1: en2[k] = sum_d emb[k,d]^2   (one wave per code row)
// ---------------------------------------------------------------------------
__global__ __launch_bounds__(256) void vq_norms(const float* __restrict__ E,
                                                float* __restrict__ en2) {
    const int w    = threadIdx.x >> 5;
    const int lane = threadIdx.x & 31;
    const int k    = blockIdx.x * 8 + w;
    const float* row = E + k * D_DIM;
    float s = 0.f;
#pragma unroll
    for (int t = 0; t < 8; ++t) {
        float v = row[lane + 32 * t];
        s += v * v;
    }
#pragma unroll
    for (int m = 16; m >= 1; m >>= 1)
        s += __shfl_xor(s, m, 32);
    if (lane == 0) en2[k] = s;
}

// ---------------------------------------------------------------------------
// Kernel 2: fused GEMM (f32 WMMA) + argmin over K.
// Block = 256 threads = 8 waves, owns 16 consecutive rows n (one batch).
// Wave w sweeps k-chunks {it*128 + w*16 .. +15}, it = 0..7  -> all 1024 codes.
// A stripe (16 rows x 256 d) lives in 128 VGPRs per wave, loaded once.
// ---------------------------------------------------------------------------
__global__ __launch_bounds__(256) void vq_argmin(const float* __restrict__ Z,
                                                 const float* __restrict__ E,
                                                 const float* __restrict__ en2,
                                                 int* __restrict__ idx) {
    __shared__ float lv[128];   // [8 waves][16 rows]
    __shared__ int   li[128];

    const int tid    = threadIdx.x;
    const int w      = tid >> 5;
    const int lane   = tid & 31;
    const int m      = lane & 15;            // row (A) / col (B) within tile
    const int koff   = (lane >> 4) << 1;     // K offset 0 or 2 per lane half
    const int n_base = blockIdx.x * 16;
    const int b      = n_base >> 10;
    const int hw     = (n_base & 1023) + m;

    // A fragments for the whole D=256 sweep: a[i] = {zf[n,4i+koff], zf[n,4i+koff+1]}
    // z_flat[n,d] = Z[b*262144 + d*1024 + hw]  (stride 1024 along d)
    v2f a[64];
    const float* ap = Z + b * BATCH_STRIDE + koff * 1024 + hw;
#pragma unroll
    for (int i = 0; i < 64; ++i) {
        a[i].x = ap[i * 4096];
        a[i].y = ap[i * 4096 + 1024];
    }

    float minv[8];
    int   mini[8];
#pragma unroll
    for (int r = 0; r < 8; ++r) { minv[r] = 3.4e38f; mini[r] = 0; }

    for (int it = 0; it < 8; ++it) {
        const int kcol = (it * 8 + w) * 16 + m;        // this lane's code column
        const float* bp = E + kcol * D_DIM + koff;     // emb row, contiguous in d
        v8f c = {0.f, 0.f, 0.f, 0.f, 0.f, 0.f, 0.f, 0.f};
#pragma unroll
        for (int i = 0; i < 64; ++i) {
            v2f bf = *(const v2f*)(bp + 4 * i);        // 8-byte aligned b64 load
            c = __builtin_amdgcn_wmma_f32_16x16x4_f32(
                    false, a[i], false, bf, (short)0, c, false, false);
        }
        const float e2 = en2[kcol];
#pragma unroll
        for (int r = 0; r < 8; ++r) {
            float s = e2 - 2.0f * c[r];                // ||e||^2 - 2 z.e
            if (s < minv[r]) { minv[r] = s; mini[r] = kcol; }
        }
    }

    // Reduce across the 16 lanes that share each row (xor masks stay in-half).
#pragma unroll
    for (int r = 0; r < 8; ++r) {
#pragma unroll
        for (int msk = 8; msk >= 1; msk >>= 1) {
            float ov = __shfl_xor(minv[r], msk, 32);
            int   oi = __shfl_xor(mini[r], msk, 32);
            if (ov < minv[r] || (ov == minv[r] && oi < mini[r])) {
                minv[r] = ov; mini[r] = oi;
            }
        }
    }
    if (m == 0) {                                      // lanes 0 and 16
        const int rowbase = (lane >> 4) * 8;           // rows 0..7 / 8..15
#pragma unroll
        for (int r = 0; r < 8; ++r) {
            lv[w * 16 + rowbase + r] = minv[r];
            li[w * 16 + rowbase + r] = mini[r];
        }
    }
    __syncthreads();
    if (tid < 16) {                                    // combine the 8 waves
        float bv = lv[tid];
        int   bi = li[tid];
#pragma unroll
        for (int ww = 1; ww < 8; ++ww) {
            float ov = lv[ww * 16 + tid];
            int   oi = li[ww * 16 + tid];
            if (ov < bv || (ov == bv && oi < bi)) { bv = ov; bi = oi; }
        }
        idx[n_base + tid] = bi;
    }
}

// ---------------------------------------------------------------------------
// Kernel 3: z_q = emb[idx] written back to [B,D,H,W] + per-block partial MSE.
// ---------------------------------------------------------------------------
__global__ __launch_bounds__(256) void vq_gather(const float* __restrict__ Z,
                                                 const float* __restrict__ E,
                                                 const int* __restrict__ idx,
                                                 float* __restrict__ out,
                                                 float* __restrict__ part) {
    __shared__ float red[256];
    const int tid    = threadIdx.x;
    const int m      = tid & 15;
    const int dg     = tid >> 4;              // 0..15
    const int n_base = blockIdx.x * 16;
    const int b      = n_base >> 10;
    const int hw     = (n_base & 1023) + m;
    const int id     = idx[n_base + m];

    const float* er   = E + id * D_DIM;
    const float* zr   = Z   + b * BATCH_STRIDE + hw;
    float*       orow = out + b * BATCH_STRIDE + hw;

    float acc = 0.f;
#pragma unroll
    for (int j = 0; j < 16; ++j) {
        const int d = dg * 16 + j;
        float e  = er[d];
        float zv = zr[d * 1024];
        orow[d * 1024] = e;                   // 16 lanes -> 64B segments
        float diff = e - zv;
        acc += diff * diff;
    }
    red[tid] = acc;
    __syncthreads();
    for (int off = 128; off >= 1; off >>= 1) {
        if (tid < off) red[tid] += red[tid + off];
        __syncthreads();
    }
    if (tid == 0) part[blockIdx.x] = red[0];
}

// ---------------------------------------------------------------------------
// Kernel 4: deterministic reduction of 1024 partials -> loss = 1.25 * MSE.
// ---------------------------------------------------------------------------
__global__ __launch_bounds__(256) void vq_loss(const float* __restrict__ part,
                                               float* __restrict__ loss) {
    __shared__ float red[256];
    const int tid = threadIdx.x;
    red[tid] = part[tid] + part[tid + 256] + part[tid + 512] + part[tid + 768];
    __syncthreads();
    for (int off = 128; off >= 1; off >>= 1) {
        if (tid < off) red[tid] += red[tid + off];
        __syncthreads();
    }
    if (tid == 0) loss[0] = red[0] * (1.25f / 4194304.0f);
}

// ---------------------------------------------------------------------------
extern "C" void kernel_launch(void* const* d_in, const int* in_sizes, int n_in,
                              void* d_out, int out_size, void* d_ws, size_t ws_size,
                              hipStream_t stream) {
    const float* Z = (const float*)d_in[0];   // z  [16,256,32,32]
    const float* E = (const float*)d_in[1];   // emb_w [1024,256]
    float* out = (float*)d_out;               // z_q (4194304) + loss (1)

    // Workspace layout: en2 (4KB) | idx (64KB) | partials (4KB)
    float* en2  = (float*)d_ws;
    int*   idx  = (int*)((char*)d_ws + 4096);
    float* part = (float*)((char*)d_ws + 4096 + 65536);
    float* loss = out + (size_t)N_VECS * D_DIM;

    vq_norms <<<K_CODES / 8, 256, 0, stream>>>(E, en2);
    vq_argmin<<<N_VECS / 16, 256, 0, stream>>>(Z, E, en2, idx);
    vq_gather<<<N_VECS / 16, 256, 0, stream>>>(Z, E, idx, out, part);
    vq_loss  <<<1,           256, 0, stream>>>(part, loss);
}